// DDDDepthDiff_60387240182105
// MI455X (gfx1250) — compile-verified
//
#include <hip/hip_runtime.h>
#include <math.h>

// ---------------- problem constants (from reference) ----------------
#define Hh      360
#define Ww      640
#define NPTS    (Hh * Ww)          // 230400
#define NPLANES 1000
#define NPAD    1024
#define MINPTS  5000
#define EPSv    1e-7f
#define CXc     334.081f
#define CYc     169.808f
#define FXc     460.585f
#define FYc     460.268f

typedef __attribute__((ext_vector_type(2))) float v2f;
typedef __attribute__((ext_vector_type(8))) float v8f;

// ---------------- workspace layout ----------------
struct Ws {
    float    sumX, sumY, sumZ, sumL;   // masked squared-diff accumulators
    int      cnt;                      // valid-mask count
    unsigned maxEnc;                   // monotone-encoded max(nan_cloud)
    int      icnt;                     // inlier count (best plane)
    unsigned zmaxEnc, zminEnc;         // monotone-encoded masked zc max/min
    float    belowSum, aboveSum;
    int      best;
    float    pa, pb, pc, pd;           // best plane a,b,c,d
    float    c0, c1, c2, tz;           // zc = c0*x + c1*y + c2*(z+tz)
    int      counts[NPAD];
    float    nx[NPAD], ny[NPAD], nz[NPAD], kk[NPAD];
};
// float4 pts[NPTS] follows Ws (256B aligned): {x,y,z,1} normalized cloud

// ---------------- helpers ----------------
__device__ __forceinline__ unsigned f2mono(float f) {
    unsigned u = __float_as_uint(f);
    return (u & 0x80000000u) ? ~u : (u | 0x80000000u);
}
__device__ __forceinline__ float mono2f(unsigned u) {
    return __uint_as_float((u & 0x80000000u) ? (u & 0x7FFFFFFFu) : ~u);
}
__device__ __forceinline__ float waveSumF(float v) {
    #pragma unroll
    for (int o = 16; o > 0; o >>= 1) v += __shfl_down(v, o);
    return v;
}
__device__ __forceinline__ int waveSumI(int v) {
    #pragma unroll
    for (int o = 16; o > 0; o >>= 1) v += __shfl_down(v, o);
    return v;
}
__device__ __forceinline__ float waveMaxF(float v) {
    #pragma unroll
    for (int o = 16; o > 0; o >>= 1) v = fmaxf(v, __shfl_down(v, o));
    return v;
}
__device__ __forceinline__ float waveMinF(float v) {
    #pragma unroll
    for (int o = 16; o > 0; o >>= 1) v = fminf(v, __shfl_down(v, o));
    return v;
}
__device__ __forceinline__ float threshOf(int epoch) {
    return fmaxf(0.025f - 0.001f * (float)epoch, 0.005f);
}

// ---------------- threefry2x32 (JAX-compatible) ----------------
__device__ __forceinline__ unsigned rotl32(unsigned x, int d) {
    return (x << d) | (x >> (32 - d));
}
__device__ void threefry2x32(unsigned k0, unsigned k1, unsigned x0, unsigned x1,
                             unsigned* o0, unsigned* o1) {
    const unsigned ks2 = 0x1BD11BDAu ^ k0 ^ k1;
    x0 += k0; x1 += k1;
    #define TFR(r) { x0 += x1; x1 = rotl32(x1, r); x1 ^= x0; }
    TFR(13) TFR(15) TFR(26) TFR(6)   x0 += k1;  x1 += ks2 + 1u;
    TFR(17) TFR(29) TFR(16) TFR(24)  x0 += ks2; x1 += k0  + 2u;
    TFR(13) TFR(15) TFR(26) TFR(6)   x0 += k0;  x1 += k1  + 3u;
    TFR(17) TFR(29) TFR(16) TFR(24)  x0 += k1;  x1 += ks2 + 4u;
    TFR(13) TFR(15) TFR(26) TFR(6)   x0 += ks2; x1 += k0  + 5u;
    #undef TFR
    *o0 = x0; *o1 = x1;
}
// random_bits(key, (3000,)) element g; halves pair (g, g+1500)
__device__ unsigned tf_bits3000(unsigned k0, unsigned k1, int g) {
    unsigned y0, y1;
    if (g < 1500) { threefry2x32(k0, k1, (unsigned)g, (unsigned)(g + 1500), &y0, &y1); return y0; }
    else          { threefry2x32(k0, k1, (unsigned)(g - 1500), (unsigned)g, &y0, &y1); return y1; }
}

// ---------------- kernels ----------------
__global__ void k_init(Ws* ws) {
    int t = threadIdx.x;
    if (t == 0) {
        ws->sumX = 0.f; ws->sumY = 0.f; ws->sumZ = 0.f; ws->sumL = 0.f;
        ws->cnt = 0; ws->icnt = 0;
        ws->maxEnc  = 0x007FFFFFu;  // mono(-inf)
        ws->zmaxEnc = 0x007FFFFFu;  // mono(-inf)
        ws->zminEnc = 0xFF800000u;  // mono(+inf)
        ws->belowSum = 0.f; ws->aboveSum = 0.f;
        ws->best = 0;
    }
    if (t < NPAD) ws->counts[t] = 0;
}

// point clouds, masked stats, nan_cloud (un-normalized) into pts4
__global__ __launch_bounds__(256) void k_pcd_stats(const float* __restrict__ fake,
                                                   const float* __restrict__ real,
                                                   Ws* ws, float4* pts4) {
    int i = blockIdx.x * blockDim.x + threadIdx.x;
    float sx = 0.f, sy = 0.f, sz = 0.f, sl = 0.f, mx = -INFINITY;
    int   cn = 0;
    if (i < NPTS) {
        int r = i / Ww, c = i % Ww;
        float cf = (float)c, rf = (float)r;
        float dr = real[i], df = fake[i];
        bool vR = (dr > 0.f) && (dr < 65535.f);
        bool vF = (df > 0.f) && (df < 65535.f);
        // *1000 point clouds: z' = depth, x' = depth*(c-CX)/FX, y' = depth*(r-CY)/FY
        float xr = dr * (cf - CXc) / FXc; if (xr == 0.f) xr = EPSv;
        float yr = dr * (rf - CYc) / FYc; if (yr == 0.f) yr = EPSv;
        float zr = dr;                    if (zr == 0.f) zr = EPSv;
        float xf = df * (cf - CXc) / FXc; if (xf == 0.f) xf = EPSv;
        float yf = df * (rf - CYc) / FYc; if (yf == 0.f) yf = EPSv;
        float zf = df;                    if (zf == 0.f) zf = EPSv;
        bool m = vR && vF;
        if (m) {
            float dx = xr - xf, dy = yr - yf, dz = zr - zf;
            float dl = logf(fabsf(zr)) - logf(fabsf(zf));
            sx = dx * dx; sy = dy * dy; sz = dz * dz; sl = dl * dl; cn = 1;
        }
        // nan_cloud: fake pcd, NaN (invalid) -> 0
        float ncx = vF ? xf : 0.f;
        float ncy = vF ? yf : 0.f;
        float ncz = vF ? zf : 0.f;
        pts4[i] = make_float4(ncx, ncy, ncz, 0.f);
        mx = fmaxf(fmaxf(ncx, ncy), ncz);
    }
    sx = waveSumF(sx); sy = waveSumF(sy); sz = waveSumF(sz); sl = waveSumF(sl);
    cn = waveSumI(cn); mx = waveMaxF(mx);
    if ((threadIdx.x & 31) == 0) {
        atomicAdd(&ws->sumX, sx); atomicAdd(&ws->sumY, sy);
        atomicAdd(&ws->sumZ, sz); atomicAdd(&ws->sumL, sl);
        atomicAdd(&ws->cnt, cn);
        atomicMax(&ws->maxEnc, f2mono(mx));
    }
}

// pts = nan_cloud / max(nan_cloud); w <- 1.0 (bias slot for the K=4 WMMA)
__global__ __launch_bounds__(256) void k_norm(Ws* ws, float4* pts4) {
    int i = blockIdx.x * blockDim.x + threadIdx.x;
    if (i >= NPTS) return;
    float s = 1.f / mono2f(ws->maxEnc);
    float4 p = pts4[i];
    pts4[i] = make_float4(p.x * s, p.y * s, p.z * s, 1.0f);
}

// threefry triples -> plane normals & offsets (SoA for WMMA B tiles)
__global__ __launch_bounds__(1024) void k_ransac_setup(Ws* ws, const float4* pts4) {
    int t = threadIdx.x;
    if (t >= NPAD) return;
    if (t >= NPLANES) { ws->nx[t] = 0.f; ws->ny[t] = 0.f; ws->nz[t] = 0.f; ws->kk[t] = 0.f; return; }
    // key(42) -> data (0,42); split(key,2): bits = random_bits(key,(2,2))
    unsigned a0, a1, b0, b1;
    threefry2x32(0u, 42u, 0u, 2u, &a0, &a1);
    threefry2x32(0u, 42u, 1u, 3u, &b0, &b1);
    unsigned k1a = a0, k1b = b0;   // key for higher_bits
    unsigned k2a = a1, k2b = b1;   // key for lower_bits
    const unsigned span = (unsigned)NPTS;
    unsigned mult = 65536u % span;
    mult = (mult * mult) % span;   // uint32 wraparound, JAX semantics
    int idx[3];
    #pragma unroll
    for (int j = 0; j < 3; j++) {
        int g = t * 3 + j;
        unsigned hi = tf_bits3000(k1a, k1b, g);
        unsigned lo = tf_bits3000(k2a, k2b, g);
        idx[j] = (int)((hi * mult + lo) % span);
    }
    float4 p0 = pts4[idx[0]], p1 = pts4[idx[1]], p2 = pts4[idx[2]];
    float ax = p1.x - p0.x, ay = p1.y - p0.y, az = p1.z - p0.z;
    float bx = p2.x - p0.x, by = p2.y - p0.y, bz = p2.z - p0.z;
    float cx = ay * bz - az * by;
    float cy = az * bx - ax * bz;
    float cz = ax * by - ay * bx;
    float inv = 1.f / sqrtf(cx * cx + cy * cy + cz * cz);
    cx *= inv; cy *= inv; cz *= inv;
    float k = -(cx * p1.x + cy * p1.y + cz * p1.z);
    ws->nx[t] = cx; ws->ny[t] = cy; ws->nz[t] = cz; ws->kk[t] = k;
}

// dist = [x y z 1] * [nx ny nz kk]^T via V_WMMA_F32_16X16X4_F32; inlier counting
#define MT_PER_WAVE 64
__global__ __launch_bounds__(32) void k_dist_count(Ws* ws, const float4* __restrict__ pts4,
                                                   const int* __restrict__ epoch_p) {
    int lane = threadIdx.x;          // wave32: one wave per block
    int half = lane >> 4, ln = lane & 15;
    int n = blockIdx.x * 16 + ln;    // plane (column) index, <= 1007 < NPAD
    float thresh = threshOf(*epoch_p);
    // B matrix 4x16: rows K=0..3 are nx,ny,nz,kk for 16 planes
    v2f b;
    b[0] = half ? ws->nz[n] : ws->nx[n];
    b[1] = half ? ws->kk[n] : ws->ny[n];
    int count = 0;
    int mt0 = blockIdx.y * MT_PER_WAVE;
    for (int i = 0; i < MT_PER_WAVE; i++) {
        int m = (mt0 + i) * 16 + ln;
        if (i + 1 < MT_PER_WAVE) __builtin_prefetch(&pts4[m + 16], 0, 0);
        float4 p = pts4[m];
        // A matrix 16x4: lanes 0-15 hold K=0,1 (x,y); lanes 16-31 hold K=2,3 (z,1)
        v2f a;
        a[0] = half ? p.z : p.x;
        a[1] = half ? p.w : p.y;
        v8f c = {0.f, 0.f, 0.f, 0.f, 0.f, 0.f, 0.f, 0.f};
        c = __builtin_amdgcn_wmma_f32_16x16x4_f32(
                /*neg_a=*/false, a, /*neg_b=*/false, b,
                /*c_mod=*/(short)0, c, /*reuse_a=*/false, /*reuse_b=*/false);
        #pragma unroll
        for (int j = 0; j < 8; j++)
            count += (fabsf(c[j]) <= thresh) ? 1 : 0;
    }
    if (n < NPLANES) atomicAdd(&ws->counts[n], count);
}

// argmax of score (counts>MIN_POINTS ? counts : -1), then rotation/translation setup
__global__ __launch_bounds__(1024) void k_best(Ws* ws) {
    __shared__ int ss[NPAD];
    __shared__ int si[NPAD];
    int t = threadIdx.x;
    ss[t] = (t < NPLANES) ? ((ws->counts[t] > MINPTS) ? ws->counts[t] : -1) : -2147483647;
    si[t] = t;
    __syncthreads();
    for (int s = 512; s > 0; s >>= 1) {
        if (t < s) {
            if (ss[t + s] > ss[t] || (ss[t + s] == ss[t] && si[t + s] < si[t])) {
                ss[t] = ss[t + s]; si[t] = si[t + s];
            }
        }
        __syncthreads();
    }
    if (t == 0) {
        int best = si[0];
        ws->best = best;
        float a = ws->nx[best], b = ws->ny[best], c = ws->nz[best], d = ws->kk[best];
        ws->pa = a; ws->pb = b; ws->pc = c; ws->pd = d;
        float n2    = a * a + b * b + c * c;
        float cos_t = c / sqrtf(n2);
        float sin_t = sqrtf((a * a + b * b) / n2);
        float sab   = sqrtf(a * a + b * b);
        float u1 = b / sab, u2 = -a / sab;
        // zc = ((pts+trans) @ R^T)[:,2] = sum_k p[k]*R[2][k]; R[2] = [-u2*s, u1*s, cos]
        ws->c0 = -u2 * sin_t;
        ws->c1 =  u1 * sin_t;
        ws->c2 =  cos_t;
        ws->tz =  d / c;
    }
}

// pass 1: inlier count for best plane + masked zc max/min
__global__ __launch_bounds__(256) void k_zc1(Ws* ws, const float4* __restrict__ pts4,
                                             const int* __restrict__ epoch_p) {
    int i = blockIdx.x * blockDim.x + threadIdx.x;
    float thresh = threshOf(*epoch_p);
    float zmx = -INFINITY, zmn = INFINITY;
    int cn = 0;
    if (i < NPTS) {
        float4 p = pts4[i];
        float dist = p.x * ws->pa + p.y * ws->pb + p.z * ws->pc + ws->pd;
        bool inl = fabsf(dist) <= thresh;
        float zc = ws->c0 * p.x + ws->c1 * p.y + ws->c2 * (p.z + ws->tz);
        if (inl) { cn = 1; zmx = zc; zmn = zc; }
    }
    cn = waveSumI(cn); zmx = waveMaxF(zmx); zmn = waveMinF(zmn);
    if ((threadIdx.x & 31) == 0) {
        atomicAdd(&ws->icnt, cn);
        atomicMax(&ws->zmaxEnc, f2mono(zmx));
        atomicMin(&ws->zminEnc, f2mono(zmn));
    }
}

// pass 2: sums of |zc - zmax| and |zc - zmin| over inliers
__global__ __launch_bounds__(256) void k_zc2(Ws* ws, const float4* __restrict__ pts4,
                                             const int* __restrict__ epoch_p) {
    int i = blockIdx.x * blockDim.x + threadIdx.x;
    float thresh = threshOf(*epoch_p);
    float below = 0.f, above = 0.f;
    if (i < NPTS) {
        float4 p = pts4[i];
        float dist = p.x * ws->pa + p.y * ws->pb + p.z * ws->pc + ws->pd;
        bool inl = fabsf(dist) <= thresh;
        if (inl) {
            float zc   = ws->c0 * p.x + ws->c1 * p.y + ws->c2 * (p.z + ws->tz);
            float zmax = mono2f(ws->zmaxEnc);
            float zmin = mono2f(ws->zminEnc);
            below = fabsf(zc - zmax);
            above = fabsf(zc - zmin);
        }
    }
    below = waveSumF(below); above = waveSumF(above);
    if ((threadIdx.x & 31) == 0) {
        atomicAdd(&ws->belowSum, below);
        atomicAdd(&ws->aboveSum, above);
    }
}

// scalar epilogue -> 7 outputs: {RMSE_log, lossX, lossY, lossZ, pmdg, loss17, loss_curv}
__global__ void k_final(Ws* ws, float* out) {
    float cnt = (float)(ws->cnt > 1 ? ws->cnt : 1);
    float lX = sqrtf(ws->sumX / cnt);
    float lY = sqrtf(ws->sumY / cnt);
    float lZ = sqrtf(ws->sumZ / cnt);
    float rl = 10000.f * sqrtf(ws->sumL / cnt);
    float l17 = rl * fabsf(10.f * (3.f - expf(lX) - expf(lY) - expf(lZ)));
    float icnt = (float)(ws->icnt > 1 ? ws->icnt : 1);
    float below = ws->belowSum / icnt;
    float above = ws->aboveSum / icnt;
    if (above == 0.f) above = 1e-7f;
    float pmdg = 1000.f * (above + below);
    out[0] = rl; out[1] = lX; out[2] = lY; out[3] = lZ;
    out[4] = pmdg; out[5] = l17; out[6] = l17 + pmdg;
}

// ---------------- launch ----------------
extern "C" void kernel_launch(void* const* d_in, const int* in_sizes, int n_in,
                              void* d_out, int out_size, void* d_ws, size_t ws_size,
                              hipStream_t stream) {
    const float* fake  = (const float*)d_in[0];
    const float* real  = (const float*)d_in[1];
    const int*   epoch = (const int*)d_in[2];
    (void)in_sizes; (void)n_in; (void)out_size; (void)ws_size;

    Ws* ws = (Ws*)d_ws;
    size_t hdr = (sizeof(Ws) + 255) & ~(size_t)255;
    float4* pts4 = (float4*)((char*)d_ws + hdr);
    float* out = (float*)d_out;

    k_init<<<1, 1024, 0, stream>>>(ws);
    k_pcd_stats<<<NPTS / 256, 256, 0, stream>>>(fake, real, ws, pts4);
    k_norm<<<NPTS / 256, 256, 0, stream>>>(ws, pts4);
    k_ransac_setup<<<1, 1024, 0, stream>>>(ws, pts4);
    k_dist_count<<<dim3(63, (NPTS / 16) / MT_PER_WAVE), 32, 0, stream>>>(ws, pts4, epoch);
    k_best<<<1, 1024, 0, stream>>>(ws);
    k_zc1<<<NPTS / 256, 256, 0, stream>>>(ws, pts4, epoch);
    k_zc2<<<NPTS / 256, 256, 0, stream>>>(ws, pts4, epoch);
    k_final<<<1, 1, 0, stream>>>(ws, out);
}